// RGCNConv_37177236914327
// MI455X (gfx1250) — compile-verified
//
#include <hip/hip_runtime.h>
#include <hip/hip_fp16.h>

typedef float v2f __attribute__((ext_vector_type(2)));
typedef float v8f __attribute__((ext_vector_type(8)));
typedef _Float16 v4h __attribute__((ext_vector_type(4)));

#define NNODES   50000
#define NEDGES   1600000
#define IN_CH    256
#define OUT_CH   128
#define NUM_REL  8
#define NUM_BASES 4
#define NSLICE   (NUM_REL + 1)   // 8 relations + root

// -------- kernel 1: build transposed weights ------------------------------
// wT[r][o][i] = sum_b att[r][b] * basis[b][i][o]   (r < 8)
// wT[8][o][i] = root[i][o]
// Transposed layout makes the WMMA B-fragment pair (K, K+1 at fixed column)
// a contiguous 8-byte load in the GEMM kernel.
__global__ __launch_bounds__(256)
void build_wT_kernel(const float* __restrict__ att,
                     const float* __restrict__ basis,
                     const float* __restrict__ root,
                     float* __restrict__ wT) {           // [9][128][256]
  int idx = blockIdx.x * blockDim.x + threadIdx.x;       // over 9*IN*OUT
  if (idx >= NSLICE * IN_CH * OUT_CH) return;
  int r  = idx / (IN_CH * OUT_CH);
  int io = idx % (IN_CH * OUT_CH);
  int i  = io / OUT_CH;                                  // coalesced read of basis/root
  int o  = io % OUT_CH;
  float v;
  if (r < NUM_REL) {
    v = 0.f;
#pragma unroll
    for (int b = 0; b < NUM_BASES; ++b)
      v += att[r * NUM_BASES + b] * basis[b * (IN_CH * OUT_CH) + io];
  } else {
    v = root[io];
  }
  wT[(size_t)r * OUT_CH * IN_CH + (size_t)o * IN_CH + i] = v;
}

// -------- kernel 2: h[r] = x @ w[r]  (fp32 WMMA, fp16 store) ---------------
// blockIdx.x : M-tile (16 rows), blockIdx.y : slice (0..7 = relation, 8 = root)
// 8 waves per block, wave i computes columns [16i, 16i+16)
__global__ __launch_bounds__(256)
void rgcn_gemm_kernel(const float* __restrict__ x,     // [N][256]
                      const float* __restrict__ wT,    // [9][128][256]
                      const float* __restrict__ bias,  // [128]
                      _Float16* __restrict__ h,        // [R][N][128]
                      float* __restrict__ out)         // [N][128]
{
  const int lane = threadIdx.x & 31;
  const int wave = threadIdx.x >> 5;
  const int half = lane >> 4;      // 0: lanes 0-15, 1: lanes 16-31
  const int lid  = lane & 15;

  const int row0 = blockIdx.x * 16;
  const int col0 = wave * 16;
  const int r    = blockIdx.y;     // 0..8

  const float* xrow = x  + (size_t)(row0 + lid) * IN_CH;
  const float* bcol = wT + ((size_t)r * OUT_CH + col0 + lid) * IN_CH;

  v8f acc = {};
  for (int k = 0; k < IN_CH; k += 4) {
    // A fragment: lane(l<16) -> M=l, K={k,k+1}; lane(l>=16) -> K={k+2,k+3}
    v2f a = *reinterpret_cast<const v2f*>(xrow + k + 2 * half);
    // B fragment: same K pair at fixed column -> contiguous in wT
    v2f b = *reinterpret_cast<const v2f*>(bcol + k + 2 * half);
    acc = __builtin_amdgcn_wmma_f32_16x16x4_f32(false, a, false, b,
                                                (short)0, acc, false, false);
  }

  if (r < NUM_REL) {
    // C/D layout: VGPR v -> row M = v + 8*half, col = lid
    _Float16* hp = h + ((size_t)r * NNODES + row0) * OUT_CH + col0 + lid;
#pragma unroll
    for (int v = 0; v < 8; ++v)
      hp[(size_t)(v + 8 * half) * OUT_CH] = (_Float16)acc[v];
  } else {
    float bv = bias[col0 + lid];
    float* op = out + (size_t)row0 * OUT_CH + col0 + lid;
#pragma unroll
    for (int v = 0; v < 8; ++v)
      op[(size_t)(v + 8 * half) * OUT_CH] = acc[v] + bv;
  }
}

// -------- kernel 3: scatter-add edge messages (gather hits L2) -------------
__global__ __launch_bounds__(256)
void edge_scatter_kernel(const int* __restrict__ esrc,
                         const int* __restrict__ edst,
                         const int* __restrict__ etyp,
                         const float* __restrict__ norm,
                         const _Float16* __restrict__ h,   // [R][N][128]
                         float* __restrict__ out)          // [N][128]
{
  const int lane   = threadIdx.x & 31;
  const int wid    = (blockIdx.x * blockDim.x + threadIdx.x) >> 5;
  const int nwaves = (gridDim.x * blockDim.x) >> 5;

  for (int e = wid; e < NEDGES; e += nwaves) {
    const int   s  = esrc[e];
    const int   d  = edst[e];
    const int   t  = etyp[e];
    const float nm = norm[e];

    // each lane owns 4 contiguous output channels -> one 8-byte load
    const v4h hv = *reinterpret_cast<const v4h*>(
        h + ((size_t)t * NNODES + s) * OUT_CH + lane * 4);

    float* op = out + (size_t)d * OUT_CH + lane * 4;
    atomicAdd(op + 0, (float)hv.x * nm);
    atomicAdd(op + 1, (float)hv.y * nm);
    atomicAdd(op + 2, (float)hv.z * nm);
    atomicAdd(op + 3, (float)hv.w * nm);
  }
}

extern "C" void kernel_launch(void* const* d_in, const int* in_sizes, int n_in,
                              void* d_out, int out_size, void* d_ws, size_t ws_size,
                              hipStream_t stream) {
  const float* x     = (const float*)d_in[0];
  const int*   esrc  = (const int*)  d_in[1];
  const int*   edst  = (const int*)  d_in[2];
  const int*   etyp  = (const int*)  d_in[3];
  const float* norm  = (const float*)d_in[4];
  const float* basis = (const float*)d_in[5];
  const float* att   = (const float*)d_in[6];
  const float* root  = (const float*)d_in[7];
  const float* bias  = (const float*)d_in[8];
  float* out = (float*)d_out;

  // workspace: wT fp32 [9][128][256] (1.18 MB), then h fp16 [8][50000][128] (102.4 MB)
  float*    wT = (float*)d_ws;
  _Float16* h  = (_Float16*)((char*)d_ws +
                             (size_t)NSLICE * IN_CH * OUT_CH * sizeof(float));

  build_wT_kernel<<<(NSLICE * IN_CH * OUT_CH + 255) / 256, 256, 0, stream>>>(
      att, basis, root, wT);

  // 3125 M-tiles x 9 slices (slices 0-7 -> h, slice 8 -> x@root + bias into d_out)
  rgcn_gemm_kernel<<<dim3(NNODES / 16, NSLICE), 256, 0, stream>>>(
      x, wT, bias, h, out);

  // runs after d_out fully initialized by the r==8 slice (same stream)
  edge_scatter_kernel<<<4096, 256, 0, stream>>>(esrc, edst, etyp, norm, h, out);
}